// Dcls3d_91156385890467
// MI455X (gfx1250) — compile-verified
//
#include <hip/hip_runtime.h>

typedef __attribute__((ext_vector_type(2))) float v2f;
typedef __attribute__((ext_vector_type(8))) float v8f;

#define OC 64
#define IC 32
#define DD 16
#define HH 32
#define WW 32
#define NB 2
#define KPC 128                      // per-channel K padded 125 -> 128
#define KDIM (IC * KPC)              // 4096
#define CCH 4                        // channels staged per LDS chunk
#define CHUNKS (IC / CCH)            // 8
#define KCHUNK (CCH * KPC)           // 512
#define TILE_W 16
#define XT_ELEMS (CCH * 5 * 5 * 20)  // 2000 floats per chunk (halo 20 wide)
#define XT_STRIDE (XT_ELEMS + 16)    // +16: per-wave zero slot for K padding

// ---------------- Phase 0: zero the constructed-kernel matrix ----------------
__global__ void zero_kmat(float* kmat, int n) {
    int i = blockIdx.x * 256 + threadIdx.x;
    if (i < n) kmat[i] = 0.0f;
}

// ---------------- Phase 1: trilinear scatter of learnable positions ---------
// kmat layout: row-major [oc][ic*128], entry (c*128 + l*25 + j*5 + i);
// indices 125..127 per channel stay zero (K padding for the 16x16x4 GEMM).
__global__ void build_kmat(const float* __restrict__ W,
                           const float* __restrict__ P,
                           float* __restrict__ kmat) {
    int t = blockIdx.x * 256 + threadIdx.x;
    const int S = OC * IC * 16;
    if (t >= S) return;
    int c = (t >> 4) & 31;
    int o = t >> 9;
    float w = W[t]; // (o,c,k) row-major == flat t

    // depth <- P[0], width <- P[1], height <- P[2]   (reference's axis twist)
    float p1 = P[0 * S + t] + 2.0f;
    float p2 = P[1 * S + t] + 2.0f;
    float p3 = P[2 * S + t] + 2.0f;
    float f1 = floorf(p1), f2 = floorf(p2), f3 = floorf(p3);
    float r1 = p1 - f1, r2 = p2 - f2, r3 = p3 - f3;
    int l0 = (int)f1, i0 = (int)f2, j0 = (int)f3;
    float wl[2] = {1.0f - r1, r1};
    float wi[2] = {1.0f - r2, r2};
    float wj[2] = {1.0f - r3, r3};

    float* row = kmat + (size_t)o * KDIM + c * KPC;
    #pragma unroll
    for (int dl = 0; dl < 2; ++dl) {
        int l = l0 + dl;
        if ((unsigned)l >= 5u) continue;
        #pragma unroll
        for (int dj = 0; dj < 2; ++dj) {
            int j = j0 + dj;
            if ((unsigned)j >= 5u) continue;
            #pragma unroll
            for (int di = 0; di < 2; ++di) {
                int i = i0 + di;
                if ((unsigned)i >= 5u) continue;
                atomicAdd(row + l * 25 + j * 5 + i, w * wl[dl] * wj[dj] * wi[di]);
            }
        }
    }
}

// ---------------- Phase 2: implicit-GEMM conv on f32 WMMA -------------------
// Block: 128 threads = 4 waves. Each WAVE owns one 16-pixel strip and ALL 64
// oc rows (4 M-tiles -> 4 accumulators), so one B gather feeds 4 WMMAs.
// Per-wave LDS x-tile; same-wave DS ops are in-order -> no per-chunk barriers.
__global__ __launch_bounds__(128) void dcls_conv(
    const float* __restrict__ x, const float* __restrict__ kmat,
    const float* __restrict__ bias, float* __restrict__ out) {
    __shared__ float ldsX[4 * XT_STRIDE]; // 4 per-wave tiles (+zero slots)
    __shared__ int ldsTab[KCHUNK];        // flat-K -> byte offset in tile

    const int tid = threadIdx.x;
    const int lane = tid & 31;
    const int wave = tid >> 5;
    const int sid = blockIdx.x * 4 + wave; // strip id: 2048 strips total
    const int w0 = (sid & 1) * TILE_W;
    const int h = (sid >> 1) & 31;
    const int d = (sid >> 6) & 15;
    const int n = sid >> 10;

    // Shared K-index -> tile-byte-offset table (identical for every chunk).
    for (int t = tid; t < KCHUNK; t += 128) {
        int cl = t >> 7;
        int r = t & 127;
        int off;
        if (r < 125) {
            int l = r / 25;
            int rr = r - l * 25;
            int j = rr / 5;
            int i = rr - j * 5;
            off = ((cl * 5 + l) * 5 + j) * 20 + i;
        } else {
            off = XT_ELEMS; // padded K lanes read the zero slot
        }
        ldsTab[t] = off * 4; // byte offset
    }
    float* tile = ldsX + wave * XT_STRIDE;
    if (lane < 16) tile[XT_ELEMS + lane] = 0.0f;
    __syncthreads(); // protects ldsTab only; tiles are wave-private

    const int laneHi = lane >> 4; // lanes 16-31 carry K=2,3
    const int p = lane & 15;      // pixel (B col) == M-row within A tile
    const char* bBase = (const char*)(tile + p);
    const float* aBase = kmat + (size_t)p * KDIM + 2 * laneHi;

    v8f acc[4] = {};

    for (int ch = 0; ch < CHUNKS; ++ch) {
        // Wave-private halo load: 4 ch x 5(d) x 5(h) x 20(w), zero padded.
        const int c0 = ch * CCH;
        for (int e = lane; e < XT_ELEMS; e += 32) {
            int cl = e / 500;
            int r = e - cl * 500;
            int dz = r / 100;
            int r2 = r - dz * 100;
            int hy = r2 / 20;
            int wx = r2 - hy * 20;
            int gd = d + dz - 2;
            int gh = h + hy - 2;
            int gw = w0 + wx - 2;
            float v = 0.0f;
            if ((unsigned)gd < (unsigned)DD && (unsigned)gh < (unsigned)HH &&
                (unsigned)gw < (unsigned)WW)
                v = x[(((size_t)n * IC + (c0 + cl)) * DD + gd) * (HH * WW) +
                      gh * WW + gw];
            tile[e] = v;
        }
        // Same-wave LDS ordering guarantees staging completes before gathers.
        const float* aP = aBase + ch * KCHUNK;
        #pragma unroll 4
        for (int s = 0; s < KCHUNK / 4; ++s) {
            int kk = 4 * s + 2 * laneHi;
            int o0 = ldsTab[kk];
            int o1 = ldsTab[kk + 1];
            v2f b;
            b.x = *(const float*)(bBase + o0); // B: 4x16 im2col tile
            b.y = *(const float*)(bBase + o1);
            #pragma unroll
            for (int r = 0; r < 4; ++r) {
                // A rows 16r+p; r-offset = r*256KB -> 24-bit imm offset
                v2f a = *(const v2f*)(aP + (size_t)r * 16 * KDIM + 4 * s);
                acc[r] = __builtin_amdgcn_wmma_f32_16x16x4_f32(
                    false, a, false, b, (short)0, acc[r], false, false);
            }
        }
    }

    // Epilogue: C/D layout -> VGPR q holds M=q (lanes 0-15) / M=q+8 (16-31).
    const size_t outPix = (size_t)d * (HH * WW) + h * WW + (w0 + p);
    #pragma unroll
    for (int r = 0; r < 4; ++r) {
        #pragma unroll
        for (int q = 0; q < 8; ++q) {
            int oc = 16 * r + 8 * laneHi + q;
            out[((size_t)n * OC + oc) * (DD * HH * WW) + outPix] =
                acc[r][q] + bias[oc];
        }
    }
}

extern "C" void kernel_launch(void* const* d_in, const int* in_sizes, int n_in,
                              void* d_out, int out_size, void* d_ws, size_t ws_size,
                              hipStream_t stream) {
    const float* x = (const float*)d_in[0];    // (2,32,16,32,32)
    const float* w = (const float*)d_in[1];    // (64,32,16)
    const float* P = (const float*)d_in[2];    // (3,64,32,16)
    const float* bias = (const float*)d_in[3]; // (64,)
    float* out = (float*)d_out;                // (2,64,16,32,32)
    float* kmat = (float*)d_ws;                // 64*4096 f32 = 1 MB scratch

    const int nz = OC * KDIM;
    zero_kmat<<<(nz + 255) / 256, 256, 0, stream>>>(kmat, nz);
    build_kmat<<<(OC * IC * 16 + 255) / 256, 256, 0, stream>>>(w, P, kmat);
    // 2048 strips / 4 waves per block = 512 blocks of 128 threads.
    dcls_conv<<<NB * DD * HH * (WW / TILE_W) / 4, 128, 0, stream>>>(x, kmat,
                                                                   bias, out);
}